// Model_24747601559836
// MI455X (gfx1250) — compile-verified
//
#include <hip/hip_runtime.h>
#include <hip/hip_bf16.h>

typedef __attribute__((ext_vector_type(2))) float v2f;
typedef __attribute__((ext_vector_type(8))) float v8f;

#define NN   69     // nodes
#define CC   69     // channels (in == out)
#define EE   2208   // edges
#define RR   3      // relations
#define BS   3      // bases
#define NPAD 288    // 4*69 = 276 output cols (3 bases + root), padded to 16*18
#define KPAD 72     // K = 69 padded to multiple of 4

// ---------------------------------------------------------------------------
// Kernel 1 (tiny, one block): build
//   - Wp (72 x 288): [basis0 | basis1 | basis2 | root] zero-padded
//   - coeff[e,b] = sum_r edge_attr[e,r]*att[r,b]
//   - CSR of edges sorted by dst with STABLE (deterministic) order
// ---------------------------------------------------------------------------
__global__ __launch_bounds__(256) void prep_kernel(
    const float* __restrict__ edge_attr, const int* __restrict__ edge_index,
    const float* __restrict__ basis, const float* __restrict__ att,
    const float* __restrict__ root,
    float* __restrict__ Wp, int* __restrict__ offs, int* __restrict__ srcS,
    float* __restrict__ coeffS)
{
  __shared__ int   s_dst[EE];
  __shared__ int   s_cnt[NN];
  __shared__ int   s_off[NN + 1];
  __shared__ float s_att[RR * BS];
  const int tid = threadIdx.x;

  // Wp: rows k in [0,72), cols j in [0,288)
  for (int idx = tid; idx < KPAD * NPAD; idx += 256) {
    int k = idx / NPAD, j = idx % NPAD;
    float v = 0.f;
    if (k < CC) {
      if (j < 3 * CC)      { int b = j / CC, o = j % CC; v = basis[(b * CC + k) * CC + o]; }
      else if (j < 4 * CC) { int o = j - 3 * CC;          v = root[k * CC + o]; }
    }
    Wp[idx] = v;
  }
  if (tid < RR * BS) s_att[tid] = att[tid];
  if (tid < NN)      s_cnt[tid] = 0;
  __syncthreads();

  for (int e = tid; e < EE; e += 256) {
    int d = edge_index[EE + e];           // dst row of edge_index
    s_dst[e] = d;
    atomicAdd(&s_cnt[d], 1);
  }
  __syncthreads();

  if (tid == 0) {
    int acc = 0;
    for (int n = 0; n < NN; ++n) { s_off[n] = acc; acc += s_cnt[n]; }
    s_off[NN] = acc;
    for (int n = 0; n <= NN; ++n) offs[n] = s_off[n];
  }
  __syncthreads();

  // stable counting-sort placement: thread n scans edges in order (deterministic)
  if (tid < NN) {
    const int n = tid;
    int pos = s_off[n];
    for (int e = 0; e < EE; ++e) {
      if (s_dst[e] == n) {
        srcS[pos] = edge_index[e];        // src row
        float a0 = edge_attr[e * RR + 0];
        float a1 = edge_attr[e * RR + 1];
        float a2 = edge_attr[e * RR + 2];
#pragma unroll
        for (int b = 0; b < BS; ++b)
          coeffS[pos * BS + b] =
              a0 * s_att[0 * BS + b] + a1 * s_att[1 * BS + b] + a2 * s_att[2 * BS + b];
        ++pos;
      }
    }
  }
}

// ---------------------------------------------------------------------------
// Kernel 2: Y(Mc x 288) = X(Mc x 69) @ Wp(72 x 288) via V_WMMA_F32_16X16X4_F32
// 4 waves per block, one 16x16 tile per wave, K swept in steps of 4.
// ---------------------------------------------------------------------------
__global__ __launch_bounds__(128) void gemm_kernel(
    const float* __restrict__ X, const float* __restrict__ Wp,
    float* __restrict__ Y, int Mc)
{
  const int wave = threadIdx.x >> 5;
  const int lane = threadIdx.x & 31;
  const int tile = blockIdx.x * 4 + wave;
  const int nTilesN = NPAD / 16;            // 18
  const int mTile = tile / nTilesN;
  const int nTile = tile % nTilesN;
  if (mTile >= (Mc >> 4)) return;           // uniform across the wave

  const int m0 = mTile << 4, n0 = nTile << 4;
  const int half = lane >> 4;               // 0: K{+0,+1}, 1: K{+2,+3}
  const int l    = lane & 15;

  const float* xrow = X + (size_t)(m0 + l) * CC;   // A: row = m0+l
  const float* wcol = Wp + (n0 + l);               // B: col = n0+l

  v8f acc = {};
  for (int k = 0; k < CC; k += 4) {                // 18 steps (K padded to 72)
    const int ka = k + 2 * half;
    v2f a, b;
    a.x = (ka     < CC) ? xrow[ka]     : 0.f;
    a.y = (ka + 1 < CC) ? xrow[ka + 1] : 0.f;
    b.x = wcol[(size_t)(ka)     * NPAD];           // rows 69..71 are zero-padded
    b.y = wcol[(size_t)(ka + 1) * NPAD];
    acc = __builtin_amdgcn_wmma_f32_16x16x4_f32(
        /*neg_a=*/false, a, /*neg_b=*/false, b,
        /*c_mod=*/(short)0, acc, /*reuse_a=*/false, /*reuse_b=*/false);
  }

  // C/D layout: VGPR v -> M = v (lanes 0-15) / M = 8+v (lanes 16-31), N = l
  float* yout = Y + (size_t)(m0 + 8 * half) * NPAD + n0 + l;
#pragma unroll
  for (int v = 0; v < 8; ++v) yout[(size_t)v * NPAD] = acc[v];
}

// ---------------------------------------------------------------------------
// Kernel 3: out[b,n,c] = Y[b,n, 207+c] + sum_{e: dst=n} sum_k cf[e,k]*Y[b,src_e, k*69+c]
// One block per (node, batch-row-tile). Edge list cached in LDS.
// Y chunk is L2-resident -> all gathers are L2 hits.
// ---------------------------------------------------------------------------
__global__ __launch_bounds__(256) void aggr_kernel(
    const float* __restrict__ Y, const float* __restrict__ coeffS,
    const int* __restrict__ srcS, const int* __restrict__ offs,
    float* __restrict__ out, int b0, int rowsPerBlock)
{
  __shared__ int   s_src[EE];
  __shared__ float s_cf[EE * 3];

  const int n  = blockIdx.x;
  const int o0 = offs[n];
  const int deg = offs[n + 1] - o0;

  for (int i = threadIdx.x; i < deg; i += 256) {
    s_src[i]        = srcS[o0 + i];
    s_cf[i * 3 + 0] = coeffS[(o0 + i) * 3 + 0];
    s_cf[i * 3 + 1] = coeffS[(o0 + i) * 3 + 1];
    s_cf[i * 3 + 2] = coeffS[(o0 + i) * 3 + 2];
  }
  __syncthreads();

  const int rowStart = blockIdx.y * rowsPerBlock;
  const int total = rowsPerBlock * CC;
  for (int idx = threadIdx.x; idx < total; idx += 256) {
    const int br = idx / CC, c = idx % CC;
    const int grow = rowStart + br;                       // chunk-local batch row
    const float* ybase = Y + (size_t)grow * NN * NPAD + c;
    float acc = ybase[(size_t)n * NPAD + 3 * CC];         // root slice (col 207+c)
    for (int e = 0; e < deg; ++e) {
      const float* yr = ybase + (size_t)s_src[e] * NPAD;
      acc = fmaf(s_cf[e * 3 + 0], yr[0],
            fmaf(s_cf[e * 3 + 1], yr[CC],
            fmaf(s_cf[e * 3 + 2], yr[2 * CC], acc)));
    }
    out[(((size_t)(b0 + grow)) * NN + n) * CC + c] = acc;
  }
}

// ---------------------------------------------------------------------------
extern "C" void kernel_launch(void* const* d_in, const int* in_sizes, int n_in,
                              void* d_out, int out_size, void* d_ws, size_t ws_size,
                              hipStream_t stream)
{
  const float* x     = (const float*)d_in[0];   // (2048, 69, 69)
  const int*   ei    = (const int*)d_in[1];     // (2, 2208)
  const float* ea    = (const float*)d_in[2];   // (2208, 3)
  const float* basis = (const float*)d_in[3];   // (3, 69, 69)
  const float* att   = (const float*)d_in[4];   // (3, 3)
  const float* root  = (const float*)d_in[5];   // (69, 69)
  float* out = (float*)d_out;

  const int Bt = 2048;

  // workspace carve-out (256B aligned)
  char* p = (char*)d_ws;
  auto carve = [&](size_t bytes) -> char* {
    char* r = p; p += (bytes + 255) & ~(size_t)255; return r;
  };
  int*   offs   = (int*)  carve((NN + 1) * sizeof(int));
  int*   srcS   = (int*)  carve(EE * sizeof(int));
  float* coeffS = (float*)carve(EE * 3 * sizeof(float));
  float* Wp     = (float*)carve((size_t)KPAD * NPAD * sizeof(float));
  const size_t fixed = (size_t)(p - (char*)d_ws);

  // largest batch chunk whose Y fits in workspace (also keeps Y L2-resident)
  int chunk = Bt;
  while (chunk > 16 && fixed + (size_t)chunk * NN * NPAD * sizeof(float) > ws_size)
    chunk >>= 1;
  float* Y = (float*)carve((size_t)chunk * NN * NPAD * sizeof(float));

  prep_kernel<<<1, 256, 0, stream>>>(ea, ei, basis, att, root, Wp, offs, srcS, coeffS);

  const int rowsPerBlock = chunk < 64 ? chunk : 64;
  for (int b0 = 0; b0 < Bt; b0 += chunk) {
    const int Mc = chunk * NN;                          // rows in this GEMM
    const int tiles = (Mc / 16) * (NPAD / 16);
    gemm_kernel<<<(tiles + 3) / 4, 128, 0, stream>>>(
        x + (size_t)b0 * NN * CC, Wp, Y, Mc);
    aggr_kernel<<<dim3(NN, chunk / rowsPerBlock), 256, 0, stream>>>(
        Y, coeffS, srcS, offs, out, b0, rowsPerBlock);
  }
}